// LengthRegulator_55611236549511
// MI455X (gfx1250) — compile-verified
//
#include <hip/hip_runtime.h>
#include <cstdint>
#include <cstddef>

// Problem constants (fixed by the reference):
#define B_  32
#define S_  512
#define C_  384      // 96 float4 per row
#define ML_ 4096

typedef float f4 __attribute__((ext_vector_type(4)));

// gfx1250 async global->LDS path (compile-guarded; falls back to plain LDS staging)
#if __has_builtin(__builtin_amdgcn_global_load_async_to_lds_b32) && \
    __has_builtin(__builtin_amdgcn_s_wait_asynccnt)
#define USE_ASYNC_LDS 1
#endif

// ---------------------------------------------------------------------------
// Kernel: expansion (placed first so the disasm snippet shows it).
// Each 256-thread block (8 waves) handles 32 time steps of one batch.
// ends[] staged in LDS via gfx1250 async global->LDS DMA; per-t uniform
// binary search in LDS; row copy = 3 x float4 per lane with NT stores.
// ---------------------------------------------------------------------------
__global__ __launch_bounds__(256) void lr_expand(const float* __restrict__ x,
                                                 const int* __restrict__ ends,
                                                 float* __restrict__ out,
                                                 float* __restrict__ mask_out) {
  __shared__ int s_ends[S_];

  constexpr int TT = 32;                 // time steps per block
  const int blocksPerB = ML_ / TT;       // 128
  const int b   = blockIdx.x / blocksPerB;
  const int t0  = (blockIdx.x % blocksPerB) * TT;
  const int tid = threadIdx.x;

  const int* gE = ends + b * S_;

#if defined(USE_ASYNC_LDS)
  {
    // Async DMA of 2 KB of ends[] into LDS: two b32 transfers per lane.
    // Builtin signature (per hipcc diagnostic): (global int* src, lds int* dst,
    // imm offset, imm cpol).
    __builtin_amdgcn_global_load_async_to_lds_b32((int*)(gE + tid),
                                                  &s_ends[tid], 0, 0);
    __builtin_amdgcn_global_load_async_to_lds_b32((int*)(gE + tid + 256),
                                                  &s_ends[tid + 256], 0, 0);
    __builtin_amdgcn_s_wait_asynccnt(0);
  }
#else
  s_ends[tid]       = gE[tid];
  s_ends[tid + 256] = gE[tid + 256];
#endif
  __syncthreads();

  const int total = s_ends[S_ - 1];
  const int wave  = tid >> 5;
  const int lane  = tid & 31;

#pragma unroll
  for (int k = 0; k < TT / 8; ++k) {     // 4 time steps per wave
    const int t = t0 + wave * 4 + k;

    // searchsorted(ends, t, side='right'): first i with ends[i] > t
    int lo = 0, hi = S_;
    while (lo < hi) {
      int mid = (lo + hi) >> 1;
      if (s_ends[mid] <= t) lo = mid + 1; else hi = mid;
    }
    int idx = lo > (S_ - 1) ? (S_ - 1) : lo;
    const bool masked = (t >= total);

    const f4* __restrict__ src = (const f4*)(x  + ((size_t)b * S_  + idx) * C_);
    f4* __restrict__       dst = (f4*)(out + ((size_t)b * ML_ + t) * C_);

    if (!masked) __builtin_prefetch((const void*)src, 0, 1);  // global_prefetch_b8

#pragma unroll
    for (int j = 0; j < 3; ++j) {        // 96 float4 / 32 lanes
      f4 v = {0.f, 0.f, 0.f, 0.f};
      if (!masked) v = src[lane + 32 * j];
      // NT: output is streamed once, never re-read -> keep x resident in cache
      __builtin_nontemporal_store(v, &dst[lane + 32 * j]);
    }
    if (lane == 0) mask_out[(size_t)b * ML_ + t] = masked ? 1.0f : 0.0f;
  }
}

// ---------------------------------------------------------------------------
// Kernel: per-batch inclusive scan of durations, clipped to max_length.
// One 512-thread block per batch; Hillis-Steele scan in LDS (9 steps).
// ---------------------------------------------------------------------------
__global__ __launch_bounds__(S_) void lr_scan(const int* __restrict__ dur,
                                              const int* __restrict__ max_len_p,
                                              int* __restrict__ ends) {
  __shared__ int sh[S_];
  const int b  = blockIdx.x;
  const int i  = threadIdx.x;
  const int ml = max_len_p[0];

  sh[i] = dur[b * S_ + i];
  __syncthreads();
#pragma unroll
  for (int off = 1; off < S_; off <<= 1) {
    int v = (i >= off) ? sh[i - off] : 0;
    __syncthreads();
    sh[i] += v;
    __syncthreads();
  }
  int e = sh[i];
  e = e < 0 ? 0 : (e > ml ? ml : e);
  ends[b * S_ + i] = e;
}

// ---------------------------------------------------------------------------
// Host launcher
// ---------------------------------------------------------------------------
extern "C" void kernel_launch(void* const* d_in, const int* in_sizes, int n_in,
                              void* d_out, int out_size, void* d_ws, size_t ws_size,
                              hipStream_t stream) {
  const float* x      = (const float*)d_in[0];
  const int*   dur    = (const int*)d_in[1];
  const int*   maxlen = (const int*)d_in[2];   // single-element int array

  float* out  = (float*)d_out;
  float* mask = out + (size_t)B_ * ML_ * C_;   // tuple outputs concatenated flat
  int*   ends = (int*)d_ws;                    // 32*512*4 = 64 KB scratch

  lr_scan<<<B_, S_, 0, stream>>>(dur, maxlen, ends);
  lr_expand<<<B_ * (ML_ / 32), 256, 0, stream>>>(x, ends, out, mask);
}